// Attention_10866267259253
// MI455X (gfx1250) — compile-verified
//
#include <hip/hip_runtime.h>
#include <math.h>

// ---- problem constants (match reference) ----
#define NUM_SEQS  16
#define NUM_HEADS 32
#define NUM_KV    8
#define HEAD      128
#define BS        16
#define MAXB      256
#define GROUP     4        // NUM_HEADS / NUM_KV
#define NWAVE     8        // waves per workgroup (wave32)

// ---- LDS layout (floats) ----
#define KSTRIDE 132        // padded row stride for Q/K/V tiles (bank-conflict-free, 8B aligned frags)
#define PSTRIDE 20         // padded row stride for P tile (q-major)
#define REDW    544        // per-wave reduce region: 4*KSTRIDE O-partial + m/l
#define QS_OFF  0
#define QS_SZ   (16 * KSTRIDE)
#define W_SZ    (2 * 16 * KSTRIDE + 16 * PSTRIDE)   // Ks + Vs + Ps per wave
#define WS_OFF  QS_SZ
#define RED_OFF (WS_OFF + NWAVE * W_SZ)
#define LDS_FLOATS (RED_OFF + NWAVE * REDW)          // 42816 floats = 171264 B (< 320 KB WGP LDS)

typedef float v2f __attribute__((ext_vector_type(2)));
typedef float v8f __attribute__((ext_vector_type(8)));

__global__ void __launch_bounds__(256, 1)
paged_attn_gqa_wmma(const float* __restrict__ q,
                    const float* __restrict__ knew,
                    const float* __restrict__ vnew,
                    const float* __restrict__ kcache,
                    const float* __restrict__ vcache,
                    const int*   __restrict__ btab,
                    const int*   __restrict__ slen,
                    float*       __restrict__ out)
{
    extern __shared__ float lds[];

    const int tid  = threadIdx.x;
    const int lane = tid & 31;
    const int wave = tid >> 5;
    const int hi   = lane >> 4;      // which half of the wave (K-dim split in WMMA layouts)
    const int lx   = lane & 15;      // row/col index within WMMA tile

    const int sidx = blockIdx.x >> 3;   // sequence
    const int kvh  = blockIdx.x & 7;    // kv head

    const int   Lq    = slen[sidx];
    const int   last  = Lq - 1;                 // position of the new decode token
    const int   nblk  = (Lq + BS - 1) >> 4;
    const float scale = 0.08838834764831845f;   // 1/sqrt(128), folded into Q

    // ---- stage scaled Q into LDS as a 16x128 tile (rows >= GROUP zeroed) ----
    for (int idx = tid; idx < 16 * HEAD; idx += 256) {
        const int r = idx >> 7, c = idx & 127;
        float v = 0.f;
        if (r < GROUP)
            v = q[((size_t)sidx * NUM_HEADS + kvh * GROUP + r) * HEAD + c] * scale;
        lds[QS_OFF + r * KSTRIDE + c] = v;
    }
    __syncthreads();

    // hoist Q B-fragments (B of S^T = K * Q^T) into registers: one v2f per K-step
    v2f qf[32];
#pragma unroll
    for (int c = 0; c < 32; ++c)
        qf[c] = *(const v2f*)&lds[QS_OFF + lx * KSTRIDE + 4 * c + 2 * hi];

    float* Ks = &lds[WS_OFF + wave * W_SZ];
    float* Vs = Ks + 16 * KSTRIDE;
    float* Ps = Vs + 16 * KSTRIDE;

    float mrun = -__builtin_inff();
    float lrun = 0.f;
    v8f   Oacc[8] = {};   // O^T, 8 groups of 16 head-dims; q in lanes, d in VGPRs

    const size_t kvoff = (size_t)kvh * HEAD;

    for (int b = wave; b < nblk; b += NWAVE) {
        const int    blk  = btab[sidx * MAXB + b];
        const size_t base = ((size_t)blk * BS) * (NUM_KV * HEAD) + kvoff;

        // ---- stage K and V tiles: 16 coalesced 512B rows each; new-token override ----
#pragma unroll 4
        for (int i = 0; i < BS; ++i) {
            const int gt = b * BS + i;
            const float* ksrc = (gt == last)
                ? (knew + ((size_t)sidx * NUM_KV + kvh) * HEAD)
                : (kcache + base + (size_t)i * (NUM_KV * HEAD));
            const float* vsrc = (gt == last)
                ? (vnew + ((size_t)sidx * NUM_KV + kvh) * HEAD)
                : (vcache + base + (size_t)i * (NUM_KV * HEAD));
            const float4 kk = *(const float4*)(ksrc + lane * 4);
            const float4 vv = *(const float4*)(vsrc + lane * 4);
            v2f t;
            t.x = kk.x; t.y = kk.y; *(v2f*)&Ks[i * KSTRIDE + lane * 4]     = t;
            t.x = kk.z; t.y = kk.w; *(v2f*)&Ks[i * KSTRIDE + lane * 4 + 2] = t;
            t.x = vv.x; t.y = vv.y; *(v2f*)&Vs[i * KSTRIDE + lane * 4]     = t;
            t.x = vv.z; t.y = vv.w; *(v2f*)&Vs[i * KSTRIDE + lane * 4 + 2] = t;
        }
        __builtin_amdgcn_wave_barrier();   // per-wave tile; DS ops are in-order per wave

        // ---- S^T = K * Q^T : 16 tokens x 16 q-cols, fp32 WMMA over head dim ----
        v8f acc = {};
#pragma unroll
        for (int c = 0; c < 32; ++c) {
            const v2f a = *(const v2f*)&Ks[lx * KSTRIDE + 4 * c + 2 * hi];
            acc = __builtin_amdgcn_wmma_f32_16x16x4_f32(false, a, false, qf[c],
                                                        (short)0, acc, false, false);
        }

        // ---- masked online softmax for this 16-token tile ----
        float sc[8];
        float mb = -__builtin_inff();
#pragma unroll
        for (int r = 0; r < 8; ++r) {
            const int t = b * BS + r + 8 * hi;     // token index of this VGPR row
            sc[r] = (t < Lq) ? acc[r] : -1e30f;
            mb = fmaxf(mb, sc[r]);
        }
        mb = fmaxf(mb, __shfl_xor(mb, 16));        // combine the two lane halves
        const float mnew  = fmaxf(mrun, mb);
        const float alpha = __expf(mrun - mnew);
        float p[8];
        float lb = 0.f;
#pragma unroll
        for (int r = 0; r < 8; ++r) { p[r] = __expf(sc[r] - mnew); lb += p[r]; }
        lb += __shfl_xor(lb, 16);
        lrun = lrun * alpha + lb;
        mrun = mnew;

        // ---- store P (q-major) to LDS for the B-fragments of the PV product ----
#pragma unroll
        for (int j = 0; j < 4; ++j) {
            v2f t; t.x = p[2 * j]; t.y = p[2 * j + 1];
            *(v2f*)&Ps[lx * PSTRIDE + 2 * j + 8 * hi] = t;
        }
        __builtin_amdgcn_wave_barrier();

        v2f pf[4];
#pragma unroll
        for (int c = 0; c < 4; ++c)
            pf[c] = *(const v2f*)&Ps[lx * PSTRIDE + 4 * c + 2 * hi];

        // ---- O^T = alpha*O^T + V^T * P, per 16-wide head-dim group ----
#pragma unroll
        for (int g = 0; g < 8; ++g) {
            v8f o = Oacc[g];
#pragma unroll
            for (int r = 0; r < 8; ++r) o[r] *= alpha;
#pragma unroll
            for (int c = 0; c < 4; ++c) {
                v2f a;
                a.x = Vs[(4 * c + 2 * hi)     * KSTRIDE + 16 * g + lx];
                a.y = Vs[(4 * c + 2 * hi + 1) * KSTRIDE + 16 * g + lx];
                o = __builtin_amdgcn_wmma_f32_16x16x4_f32(false, a, false, pf[c],
                                                          (short)0, o, false, false);
            }
            Oacc[g] = o;
        }
        __builtin_amdgcn_wave_barrier();
    }

    // ---- write per-wave partials (unnormalized O^T plus m, l) ----
    float* red = &lds[RED_OFF + wave * REDW];
    if (lx < GROUP) {
#pragma unroll
        for (int g = 0; g < 8; ++g)
#pragma unroll
            for (int r = 0; r < 8; ++r)
                red[lx * KSTRIDE + 16 * g + r + 8 * hi] = Oacc[g][r];
        if (!hi) {
            red[4 * KSTRIDE + 2 * lx]     = mrun;
            red[4 * KSTRIDE + 2 * lx + 1] = lrun;
        }
    }
    __syncthreads();

    // ---- flash-decoding combine across the 8 waves, normalize, store ----
    for (int idx = tid; idx < GROUP * HEAD; idx += 256) {
        const int qi = idx >> 7, d = idx & 127;
        float M = -__builtin_inff();
#pragma unroll
        for (int w = 0; w < NWAVE; ++w)
            M = fmaxf(M, lds[RED_OFF + w * REDW + 4 * KSTRIDE + 2 * qi]);
        float Lsum = 0.f, val = 0.f;
#pragma unroll
        for (int w = 0; w < NWAVE; ++w) {
            const float mw = lds[RED_OFF + w * REDW + 4 * KSTRIDE + 2 * qi];
            const float lw = lds[RED_OFF + w * REDW + 4 * KSTRIDE + 2 * qi + 1];
            const float f  = (mw > -1e37f) ? __expf(mw - M) : 0.f;
            Lsum += lw * f;
            val  += lds[RED_OFF + w * REDW + qi * KSTRIDE + d] * f;
        }
        out[(size_t)sidx * (NUM_HEADS * HEAD) + (kvh * GROUP + qi) * HEAD + d] = val / Lsum;
    }
}

extern "C" void kernel_launch(void* const* d_in, const int* in_sizes, int n_in,
                              void* d_out, int out_size, void* d_ws, size_t ws_size,
                              hipStream_t stream)
{
    (void)in_sizes; (void)n_in; (void)out_size; (void)d_ws; (void)ws_size;

    const float* q  = (const float*)d_in[0];
    const float* kn = (const float*)d_in[1];
    const float* vn = (const float*)d_in[2];
    const float* kc = (const float*)d_in[3];
    const float* vc = (const float*)d_in[4];
    const int*   bt = (const int*)d_in[5];
    const int*   sl = (const int*)d_in[6];
    float*       o  = (float*)d_out;

    const size_t shmem = (size_t)LDS_FLOATS * sizeof(float);   // ~167 KB, fits 320 KB WGP LDS
    hipFuncSetAttribute((const void*)paged_attn_gqa_wmma,
                        hipFuncAttributeMaxDynamicSharedMemorySize, (int)shmem);

    dim3 grid(NUM_SEQS * NUM_KV);   // one WG per (seq, kv_head)
    dim3 block(256);                // 8 wave32s
    paged_attn_gqa_wmma<<<grid, block, shmem, stream>>>(q, kn, vn, kc, vc, bt, sl, o);
}